// DeepLatent_9689446220325
// MI455X (gfx1250) — compile-verified
//
#include <hip/hip_runtime.h>
#include <math.h>

typedef __attribute__((ext_vector_type(2))) float v2f;
typedef __attribute__((ext_vector_type(8))) float v8f;

#define BATCH 32
#define NPTS  2048
#define TILES (NPTS / 16)   // 128 column tiles of 16

// ---------------------------------------------------------------------------
// Init: zero output scalar and per-batch directional accumulators.
// ---------------------------------------------------------------------------
__global__ void dl_init_kernel(float* __restrict__ out,
                               float* __restrict__ dsum) {
    int i = blockIdx.x * blockDim.x + threadIdx.x;
    if (i == 0) out[0] = 0.0f;
    if (i < 2 * BATCH) dsum[i] = 0.0f;
}

// ---------------------------------------------------------------------------
// One-directional Chamfer: dsum[b] += sum_i min_j ||src_i - dst_j||^2.
// Called twice with (est,gt) and (gt,est) -> dist1 and dist2, no col atomics.
//
// One wave32 per (batch, 16-row src tile).
// d2[i][j] = |x_i|^2 + |y_j|^2 - 2 x_i.y_j via V_WMMA_F32_16X16X4_F32:
//   A row (16x4) = (-2x0, -2x1, -2x2, 1)   [lanes 0-15: K0,K1; 16-31: K2,K3]
//   B col (4x16) = ( y0,  y1,  y2, |y|^2)  [lanes 0-15: K0,K1; 16-31: K2,K3]
// Row mins live in registers; one shuffle-reduce + one atomicAdd per wave.
// ---------------------------------------------------------------------------
__global__ __launch_bounds__(256) void dl_chamfer_dir_kernel(
    const float* __restrict__ src, const float* __restrict__ dst,
    float* __restrict__ dsum) {

    const int lane = threadIdx.x & 31;
    const int wave = threadIdx.x >> 5;
    const int w    = blockIdx.x * 8 + wave;       // global wave id
    const int b    = w / TILES;
    const int i0   = (w % TILES) * 16;            // first src row of tile
    const int half = lane >> 4;                   // 0: lanes 0-15, 1: 16-31
    const int l16  = lane & 15;

    const float* sb = src + (size_t)b * NPTS * 3;
    const float* db = dst + (size_t)b * NPTS * 3;

    // ---- A operand (fixed per wave), ISA 32-bit A 16x4 layout.
    const int arow = i0 + l16;
    const float ax = sb[arow * 3 + 0];
    const float ay = sb[arow * 3 + 1];
    const float az = sb[arow * 3 + 2];
    v2f A;
    A.x = half ? (-2.0f * az) : (-2.0f * ax);
    A.y = half ? 1.0f         : (-2.0f * ay);

    // ---- |src row|^2 for the 8 rows this lane's D VGPRs map to
    // (C/D layout: lanes 0-15 -> M = v, lanes 16-31 -> M = v + 8).
    float x2v[8];
#pragma unroll
    for (int v = 0; v < 8; ++v) {
        const int r = i0 + half * 8 + v;
        const float ex = sb[r * 3 + 0];
        const float ey = sb[r * 3 + 1];
        const float ez = sb[r * 3 + 2];
        x2v[v] = ex * ex + ey * ey + ez * ez;
    }

    float rmin[8];
#pragma unroll
    for (int v = 0; v < 8; ++v) rmin[v] = 3.4e38f;

    // ---- software-pipelined loop over dst column tiles
    float gx = db[l16 * 3 + 0];
    float gy = db[l16 * 3 + 1];
    float gz = db[l16 * 3 + 2];

    for (int jt = 0; jt < TILES; ++jt) {
        const float cgx = gx, cgy = gy, cgz = gz;

        // prefetch next tile (clamped; uniform, branch-free)
        const int nt   = (jt + 1 < TILES) ? (jt + 1) : (TILES - 1);
        const int ncol = nt * 16 + l16;
        gx = db[ncol * 3 + 0];
        gy = db[ncol * 3 + 1];
        gz = db[ncol * 3 + 2];

        const float y2 = cgx * cgx + cgy * cgy + cgz * cgz;
        v2f Bm;
        Bm.x = half ? cgz : cgx;
        Bm.y = half ? y2  : cgy;

        v8f C = {0.f, 0.f, 0.f, 0.f, 0.f, 0.f, 0.f, 0.f};
        // (neg_a, A, neg_b, B, c_mod, C, reuse_a, reuse_b)
        C = __builtin_amdgcn_wmma_f32_16x16x4_f32(
                false, A, false, Bm, (short)0, C, false, false);

#pragma unroll
        for (int v = 0; v < 8; ++v) {
            float d = C[v] + x2v[v];        // |x|^2 + (|y|^2 - 2 x.y)
            d = fmaxf(d, 0.0f);
            rmin[v] = fminf(rmin[v], d);
        }
    }

    // ---- min over the 16 columns held across lanes of each half
#pragma unroll
    for (int v = 0; v < 8; ++v) {
#pragma unroll
        for (int m = 1; m < 16; m <<= 1)
            rmin[v] = fminf(rmin[v], __shfl_xor(rmin[v], m, 16));
    }
    float hsum = 0.0f;
#pragma unroll
    for (int v = 0; v < 8; ++v) hsum += rmin[v];       // rows 0-7 / 8-15
    const float tot = hsum + __shfl_xor(hsum, 16, 32); // all 16 rows
    if (lane == 0) atomicAdd(&dsum[b], tot);
}

// ---------------------------------------------------------------------------
// Finalize: per batch, BCE-with-logits sum + both chamfer sums -> scalar.
// ---------------------------------------------------------------------------
__global__ __launch_bounds__(256) void dl_finalize_kernel(
    const float* __restrict__ dsum1,
    const float* __restrict__ dsum2,
    const float* __restrict__ labels_est,
    const float* __restrict__ labels,
    float* __restrict__ out) {

    const int b   = blockIdx.x;
    const int tid = threadIdx.x;

    float acc = 0.0f;
    for (int j = tid; j < NPTS; j += 256) {
        const float z = labels_est[(size_t)b * NPTS + j];
        const float t = labels[(size_t)b * NPTS + j];
        // stable: max(z,0) - z*t + log(1+exp(-|z|))
        acc += fmaxf(z, 0.0f) - z * t + log1pf(expf(-fabsf(z)));
    }

    __shared__ float s[256];
    s[tid] = acc;
    __syncthreads();
    for (int o = 128; o > 0; o >>= 1) {
        if (tid < o) s[tid] += s[tid + o];
        __syncthreads();
    }
    if (tid == 0) {
        const float per = (s[0] + dsum1[b] + dsum2[b]) / (float)NPTS;
        atomicAdd(out, per / (float)BATCH);
    }
}

// ---------------------------------------------------------------------------
extern "C" void kernel_launch(void* const* d_in, const int* in_sizes, int n_in,
                              void* d_out, int out_size, void* d_ws, size_t ws_size,
                              hipStream_t stream) {
    const float* obs_est    = (const float*)d_in[0];  // [32,2048,3]
    const float* obs_gt     = (const float*)d_in[1];  // [32,2048,3]
    const float* labels_est = (const float*)d_in[2];  // [32,2048]
    const float* labels     = (const float*)d_in[3];  // [32,2048]
    float* out = (float*)d_out;                       // scalar

    float* dsum  = (float*)d_ws;      // [2*BATCH]: dist1 sums, then dist2 sums
    float* dsum1 = dsum;
    float* dsum2 = dsum + BATCH;

    dl_init_kernel<<<1, 256, 0, stream>>>(out, dsum);

    // 32 batches * 128 row tiles = 4096 waves, 8 waves per 256-thread block
    const int blocks = (BATCH * TILES) / 8;
    dl_chamfer_dir_kernel<<<blocks, 256, 0, stream>>>(obs_est, obs_gt, dsum1);
    dl_chamfer_dir_kernel<<<blocks, 256, 0, stream>>>(obs_gt, obs_est, dsum2);

    dl_finalize_kernel<<<BATCH, 256, 0, stream>>>(
        dsum1, dsum2, labels_est, labels, out);
}